// GATConvCuGraph_55920474194604
// MI455X (gfx1250) — compile-verified
//
#include <hip/hip_runtime.h>

// ---------------------------------------------------------------------------
// GATConv forward for MI455X (gfx1250, wave32).
//   h = x @ W^T            -> fp32 WMMA 16x16x4 (exact fp32 path)
//   a_src/a_dst dots       -> scalar VALU kernel
//   edge softmax + scatter -> atomicMax (ordered-uint trick) + atomicAdd,
//                             h gathers are L2-resident (51.2MB < 192MB L2)
// ---------------------------------------------------------------------------

typedef __attribute__((ext_vector_type(2))) float v2f;
typedef __attribute__((ext_vector_type(8))) float v8f;

#define NN   100000
#define NE   1600000
#define ET   (NN + NE)     // edges incl. self-loops
#define CIN  128
#define HF   128           // HEADS * OUT_CH
#define NH   8
#define FPH  16
#define NEG_SLOPE 0.2f

// monotonic float<->uint mapping for atomicMax on floats
__device__ __forceinline__ unsigned enc_f32(float f) {
  unsigned u = __float_as_uint(f);
  return (u & 0x80000000u) ? ~u : (u | 0x80000000u);
}
__device__ __forceinline__ float dec_f32(unsigned u) {
  return (u & 0x80000000u) ? __uint_as_float(u ^ 0x80000000u)
                           : __uint_as_float(~u);
}
#define ENC_NEG_INF 0x007FFFFFu   // enc_f32(-inf)

__device__ __forceinline__ void edge_nodes(int e, const long long* __restrict__ ei,
                                           int& s, int& d) {
  if (e < NE) { s = (int)ei[e]; d = (int)ei[NE + e]; }
  else        { s = d = e - NE; }   // self-loop
}

// ---------------------------------------------------------------------------
__global__ void gat_init_out(float* __restrict__ out, const float* __restrict__ bias) {
  int i = blockIdx.x * 256 + threadIdx.x;
  if (i < NN * HF) out[i] = bias[i & (HF - 1)];
}

__global__ void gat_init_mdenom(unsigned* __restrict__ m_enc, float* __restrict__ denom) {
  int i = blockIdx.x * 256 + threadIdx.x;
  if (i < NN * NH) { m_enc[i] = ENC_NEG_INF; denom[i] = 0.0f; }
}

// ---------------------------------------------------------------------------
// h[n][o] = sum_k x[n][k] * W[o][k].  One wave per 16x16 output tile.
// Block = 256 threads = 8 waves; wave w covers output cols [16w, 16w+16).
// gridDim.x = 6250 row tiles (100000 rows exactly).
// A 16x4 fp32: lane<16 -> K=0,1 ; lane>=16 -> K=2,3 of row M=lane&15.
// B 4x16 fp32: VGPR0 = rows K={0,2}, VGPR1 = rows K={1,3}, N=lane&15.
// C/D: VGPR v -> M = (lane>>4)*8 + v, N = lane&15.
__global__ void gat_gemm_wmma(const float* __restrict__ x,
                              const float* __restrict__ W,
                              float* __restrict__ h) {
  const int wave = threadIdx.x >> 5;   // output col tile (0..7)
  const int lane = threadIdx.x & 31;
  const int row  = lane & 15;
  const int half = lane >> 4;
  const int m0   = blockIdx.x * 16;

  const float* xrow = x + (size_t)(m0 + row) * CIN + half * 2;
  const float* wrow = W + (size_t)(wave * 16 + row) * CIN + half * 2;

  v8f c = {};
#pragma unroll
  for (int s = 0; s < CIN / 4; ++s) {
    v2f a = *(const v2f*)(xrow + s * 4);
    v2f b = *(const v2f*)(wrow + s * 4);
    // (neg_a, A, neg_b, B, c_mod, C, reuse_a, reuse_b)
    c = __builtin_amdgcn_wmma_f32_16x16x4_f32(false, a, false, b,
                                              (short)0, c, false, false);
  }

#pragma unroll
  for (int v = 0; v < 8; ++v)
    h[(size_t)(m0 + half * 8 + v) * HF + wave * 16 + row] = c[v];
}

// ---------------------------------------------------------------------------
// a_src[n][hd] = <h[n,hd,:], att_src[hd,:]> ; a_dst likewise.
__global__ void gat_attn_coef(const float* __restrict__ h,
                              const float* __restrict__ att,
                              float* __restrict__ a_src,
                              float* __restrict__ a_dst) {
  int i = blockIdx.x * 256 + threadIdx.x;
  if (i >= NN * NH) return;
  int n = i >> 3, hd = i & 7;
  const float* hp = h + (size_t)n * HF + hd * FPH;
  const float* as = att + hd * FPH;
  const float* ad = att + HF + hd * FPH;
  float s0 = 0.0f, s1 = 0.0f;
#pragma unroll
  for (int f = 0; f < FPH; ++f) {
    float v = hp[f];
    s0 += v * as[f];
    s1 += v * ad[f];
  }
  a_src[i] = s0;
  a_dst[i] = s1;
}

// ---------------------------------------------------------------------------
// Pass A: segment max of LeakyReLU logits over incoming edges (per head).
__global__ void gat_edge_max(const long long* __restrict__ ei,
                             const float* __restrict__ a_src,
                             const float* __restrict__ a_dst,
                             unsigned* __restrict__ m_enc) {
  int i = blockIdx.x * 256 + threadIdx.x;
  if (i >= ET * NH) return;
  int e = i >> 3, hd = i & 7;
  int s, d; edge_nodes(e, ei, s, d);
  float v = a_src[s * NH + hd] + a_dst[d * NH + hd];
  v = (v >= 0.0f) ? v : NEG_SLOPE * v;
  atomicMax(&m_enc[d * NH + hd], enc_f32(v));
}

// Pass B: denom[d][hd] = sum exp(e - m).
__global__ void gat_edge_denom(const long long* __restrict__ ei,
                               const float* __restrict__ a_src,
                               const float* __restrict__ a_dst,
                               const unsigned* __restrict__ m_enc,
                               float* __restrict__ denom) {
  int i = blockIdx.x * 256 + threadIdx.x;
  if (i >= ET * NH) return;
  int e = i >> 3, hd = i & 7;
  int s, d; edge_nodes(e, ei, s, d);
  int aidx = d * NH + hd;
  float v = a_src[s * NH + hd] + a_dst[aidx];
  v = (v >= 0.0f) ? v : NEG_SLOPE * v;
  atomicAdd(&denom[aidx], expf(v - dec_f32(m_enc[aidx])));
}

// Pass C: out[d] += alpha * h[s].  One wave per edge; lane covers 4 floats.
__global__ void gat_edge_aggregate(const long long* __restrict__ ei,
                                   const float* __restrict__ a_src,
                                   const float* __restrict__ a_dst,
                                   const unsigned* __restrict__ m_enc,
                                   const float* __restrict__ denom,
                                   const float* __restrict__ h,
                                   float* __restrict__ out) {
  unsigned gid = blockIdx.x * 256 + threadIdx.x;
  int e    = (int)(gid >> 5);
  int lane = (int)(gid & 31);
  if (e >= ET) return;
  int s, d; edge_nodes(e, ei, s, d);
  int hd   = lane >> 2;              // 4 lanes per head (16 floats / head)
  int aidx = d * NH + hd;
  float v = a_src[s * NH + hd] + a_dst[aidx];
  v = (v >= 0.0f) ? v : NEG_SLOPE * v;
  float alpha = expf(v - dec_f32(m_enc[aidx])) / denom[aidx];

  const float4 hv = *(const float4*)(h + (size_t)s * HF + lane * 4);
  float* op = out + (size_t)d * HF + lane * 4;
  atomicAdd(op + 0, alpha * hv.x);
  atomicAdd(op + 1, alpha * hv.y);
  atomicAdd(op + 2, alpha * hv.z);
  atomicAdd(op + 3, alpha * hv.w);
}

// ---------------------------------------------------------------------------
extern "C" void kernel_launch(void* const* d_in, const int* in_sizes, int n_in,
                              void* d_out, int out_size, void* d_ws, size_t ws_size,
                              hipStream_t stream) {
  const float*     x    = (const float*)d_in[0];      // [NN, CIN]
  const float*     W    = (const float*)d_in[1];      // [HF, CIN]
  const float*     att  = (const float*)d_in[2];      // [2*HF]
  const float*     bias = (const float*)d_in[3];      // [HF]
  const long long* ei   = (const long long*)d_in[4];  // [2, NE] int64
  float*           out  = (float*)d_out;              // [NN, HF]

  // workspace layout (floats): h | a_src | a_dst | m_enc | denom  (~64 MB)
  float*    ws    = (float*)d_ws;
  float*    h     = ws;                               // NN*HF  = 12.8M
  float*    a_src = h + (size_t)NN * HF;              // NN*NH  = 0.8M
  float*    a_dst = a_src + (size_t)NN * NH;
  unsigned* m_enc = (unsigned*)(a_dst + (size_t)NN * NH);
  float*    denom = (float*)(m_enc + (size_t)NN * NH);

  const int B = 256;
  gat_init_out   <<<(NN * HF + B - 1) / B, B, 0, stream>>>(out, bias);
  gat_init_mdenom<<<(NN * NH + B - 1) / B, B, 0, stream>>>(m_enc, denom);

  gat_gemm_wmma  <<<NN / 16, B, 0, stream>>>(x, W, h);          // 6250 blocks
  gat_attn_coef  <<<(NN * NH + B - 1) / B, B, 0, stream>>>(h, att, a_src, a_dst);

  gat_edge_max   <<<(ET * NH + B - 1) / B, B, 0, stream>>>(ei, a_src, a_dst, m_enc);
  gat_edge_denom <<<(ET * NH + B - 1) / B, B, 0, stream>>>(ei, a_src, a_dst, m_enc, denom);

  // one wave (32 lanes) per edge
  long long aggT = (long long)ET * 32;
  gat_edge_aggregate<<<(unsigned)((aggT + B - 1) / B), B, 0, stream>>>(
      ei, a_src, a_dst, m_enc, denom, h, out);
}